// Graph_to_Featuremaps_savemem_25357486916071
// MI455X (gfx1250) — compile-verified
//
#include <hip/hip_runtime.h>
#include <hip/hip_bf16.h>

typedef float v2f __attribute__((ext_vector_type(2)));
typedef float v4f __attribute__((ext_vector_type(4)));
typedef float v8f __attribute__((ext_vector_type(8)));

static constexpr int B_    = 16;
static constexpr int NODES = 64;
static constexpr int HID   = 128;
static constexpr int C_    = 256;
static constexpr int HW    = 128 * 128;

// ---------------------------------------------------------------------------
// Kernel A: per-batch softmax attention + node-weighted feature vector
//   s[n] = <x[b,n,:], nfh>;  a = softmax(s);  y[b,h] = sum_n a[n]*x[b,n,h]
// One block per batch element, 128 threads. Work is trivially small.
// ---------------------------------------------------------------------------
__global__ void attn_reduce_kernel(const float* __restrict__ x,   // (B,NODES,HID)
                                   const float* __restrict__ nfh, // (HID,1)
                                   float* __restrict__ y)         // (B,HID)
{
    __shared__ float s[NODES];
    __shared__ float a[NODES];
    const int b = blockIdx.x;
    const int t = threadIdx.x;            // 0..127
    const float* xb = x + (size_t)b * NODES * HID;

    if (t < NODES) {
        float acc = 0.0f;
        #pragma unroll 4
        for (int h = 0; h < HID; ++h) acc += xb[t * HID + h] * nfh[h];
        s[t] = acc;
    }
    __syncthreads();

    if (t == 0) {                          // serial softmax over 64 — negligible
        float m = s[0];
        for (int n = 1; n < NODES; ++n) m = fmaxf(m, s[n]);
        float sum = 0.0f;
        for (int n = 0; n < NODES; ++n) { float e = __expf(s[n] - m); a[n] = e; sum += e; }
        float inv = 1.0f / sum;
        for (int n = 0; n < NODES; ++n) a[n] *= inv;
    }
    __syncthreads();

    // y[b,t] — coalesced across t
    float acc = 0.0f;
    #pragma unroll 4
    for (int n = 0; n < NODES; ++n) acc += a[n] * xb[n * HID + t];
    y[b * HID + t] = acc;
}

// ---------------------------------------------------------------------------
// Kernel B: V = relu(Y @ W), Y:(16,128), W:(128,256), V:(16,256)
// One block of 512 threads = 16 wave32's. Wave w owns the 16x16 output tile
// covering columns [16w, 16w+16). K=128 consumed in 32 steps of
// V_WMMA_F32_16X16X4_F32 (full fp32 precision).
//
// ISA fragment layouts (§7.12.2):
//   A 16x4 f32: lane L: M=L&15, VGPR{0,1} = K = k0 + 2*(L>=16) + {0,1}
//   B  4x16   : lane L: N=L&15, VGPR{0,1} = K = k0 + 2*(L>=16) + {0,1}
//   C/D 16x16 : lane L: N=L&15, VGPR r    = M = r + 8*(L>=16)
// ---------------------------------------------------------------------------
__global__ void proj_wmma_kernel(const float* __restrict__ y, // (16,128)
                                 const float* __restrict__ w, // (128,256)
                                 float* __restrict__ v)       // (16,256)
{
    const int lane = threadIdx.x & 31;
    const int wave = threadIdx.x >> 5;    // 0..15 -> N tile
    const int half = lane >> 4;           // 0/1
    const int l    = lane & 15;
    const int ncol = wave * 16 + l;

    v8f c = {};
    #pragma unroll 4
    for (int k0 = 0; k0 < HID; k0 += 4) {
        const int ka = k0 + 2 * half;
        v2f af, bf;
        af.x = y[l * HID + ka + 0];
        af.y = y[l * HID + ka + 1];
        bf.x = w[(ka + 0) * C_ + ncol];
        bf.y = w[(ka + 1) * C_ + ncol];
        // (neg_a, A, neg_b, B, c_mod, C, reuse_a, reuse_b)
        c = __builtin_amdgcn_wmma_f32_16x16x4_f32(
                false, af, false, bf, (short)0, c, false, false);
    }

    #pragma unroll
    for (int r = 0; r < 8; ++r) {
        const int m = r + 8 * half;
        v[m * C_ + ncol] = fmaxf(c[r], 0.0f);
    }
}

// ---------------------------------------------------------------------------
// Kernel C: out[b,c,:,:] = v[b,c]  — the 256 MB bandwidth-bound broadcast.
// One block per (b,c) plane (4096 blocks); 256 threads x 16 float4 NT stores
// = 64 KB contiguous per block. Non-temporal: 256 MB output >> 192 MB L2.
// ---------------------------------------------------------------------------
__global__ void broadcast_kernel(const float* __restrict__ v, // (16,256)
                                 float* __restrict__ out)     // (B,C,H,W)
{
    const int plane = blockIdx.x;         // b*256 + c, scalar load (uniform)
    const float val = v[plane];
    v4f q = { val, val, val, val };
    v4f* o = (v4f*)(out + (size_t)plane * HW);
    const int t = threadIdx.x;
    #pragma unroll
    for (int i = 0; i < 16; ++i) {
        __builtin_nontemporal_store(q, o + t + i * 256);
    }
}

// ---------------------------------------------------------------------------
extern "C" void kernel_launch(void* const* d_in, const int* in_sizes, int n_in,
                              void* d_out, int out_size, void* d_ws, size_t ws_size,
                              hipStream_t stream) {
    const float* x   = (const float*)d_in[0];  // (1,B,NODES,HID)
    // d_in[1] res_feature, d_in[2] node_fea_for_res: provably unused
    // (softmax over nodes is invariant to the per-pixel additive term).
    const float* nfh = (const float*)d_in[3];  // (HID,1)
    const float* w   = (const float*)d_in[4];  // (HID,C)
    float* out = (float*)d_out;

    float* ws_y = (float*)d_ws;                       // 16*128 floats = 8 KB
    float* ws_v = (float*)((char*)d_ws + 8192);       // 16*256 floats = 16 KB

    attn_reduce_kernel<<<B_, HID, 0, stream>>>(x, nfh, ws_y);
    proj_wmma_kernel<<<1, 512, 0, stream>>>(ws_y, w, ws_v);
    broadcast_kernel<<<B_ * C_, 256, 0, stream>>>(ws_v, out);
}